// LightGCN_41291815584253
// MI455X (gfx1250) — compile-verified
//
#include <hip/hip_runtime.h>
#include <hip/hip_bf16.h>

#define D 32

typedef __attribute__((ext_vector_type(2))) float v2f;
typedef __attribute__((ext_vector_type(8))) float v8f;

// ---------------------------------------------------------------------------
// 1) zero degree histograms and the aggregation buffer (must run every call)
// ---------------------------------------------------------------------------
__global__ void gcn_init(int* __restrict__ deg_out, int* __restrict__ deg_in,
                         float* __restrict__ agg, int n_nodes, int n_agg) {
    int i = blockIdx.x * blockDim.x + threadIdx.x;
    if (i < n_agg)   agg[i] = 0.0f;
    if (i < n_nodes) { deg_out[i] = 0; deg_in[i] = 0; }
}

// ---------------------------------------------------------------------------
// 2) degree histograms: one thread per edge, two int atomics
// ---------------------------------------------------------------------------
__global__ void gcn_degree(const int* __restrict__ src, const int* __restrict__ dst,
                           int* __restrict__ deg_out, int* __restrict__ deg_in,
                           int n_edges) {
    int e = blockIdx.x * blockDim.x + threadIdx.x;
    if (e >= n_edges) return;
    atomicAdd(&deg_out[src[e]], 1);
    atomicAdd(&deg_in[dst[e]], 1);
}

// ---------------------------------------------------------------------------
// 3) norms = rsqrt(max(deg,1))
// ---------------------------------------------------------------------------
__global__ void gcn_norm(const int* __restrict__ deg_out, const int* __restrict__ deg_in,
                         float* __restrict__ out_norm, float* __restrict__ in_norm,
                         int n_nodes) {
    int i = blockIdx.x * blockDim.x + threadIdx.x;
    if (i >= n_nodes) return;
    out_norm[i] = rsqrtf(fmaxf((float)deg_out[i], 1.0f));
    in_norm[i]  = rsqrtf(fmaxf((float)deg_in[i], 1.0f));
}

// ---------------------------------------------------------------------------
// 4) edge scatter: one wave32 per edge, one lane per feature (D==32).
//    Each wave issues a contiguous 128B run of global_atomic_add_f32 into L2.
// ---------------------------------------------------------------------------
__global__ void gcn_scatter(const float* __restrict__ x,
                            const int* __restrict__ src, const int* __restrict__ dst,
                            const float* __restrict__ out_norm,
                            float* __restrict__ agg, int n_edges) {
    const int lane = threadIdx.x & 31;
    const int e = blockIdx.x * (blockDim.x >> 5) + (threadIdx.x >> 5);
    if (e >= n_edges) return;                 // wave-uniform guard
    const int s = src[e];
    const int t = dst[e];
    const float v = x[s * D + lane] * out_norm[s];
    unsafeAtomicAdd(&agg[t * D + lane], v);   // -> global_atomic_add_f32
}

// ---------------------------------------------------------------------------
// 5) out = (agg * in_norm[:,None]) @ W + b   via V_WMMA_F32_16X16X4_F32.
//    One wave -> 16 node rows x 32 output cols (two 16x16 C tiles),
//    K=32 in 8 steps of 4. in_norm fused into A-load, bias fused into store.
//
//    Fragment layouts (CDNA5 ISA, wave32):
//      A 16x4:  lanes 0-15: {A[m][k0],A[m][k0+1]},  lanes 16-31: {A[m][k0+2],A[m][k0+3]}
//      B 4x16:  vgpr0: lanes0-15=B[k0][n], lanes16-31=B[k0+2][n]; vgpr1: k0+1 / k0+3
//      C 16x16: vgpr i: lanes0-15 = C[m=i][n=lane], lanes16-31 = C[m=i+8][n=lane-16]
// ---------------------------------------------------------------------------
__global__ void gcn_gemm_wmma(const float* __restrict__ agg,
                              const float* __restrict__ in_norm,
                              const float* __restrict__ W,
                              const float* __restrict__ bias,
                              float* __restrict__ out, int n_nodes) {
    const int lane = threadIdx.x & 31;
    const int wave = blockIdx.x * (blockDim.x >> 5) + (threadIdx.x >> 5);
    const int row0 = wave * 16;
    if (row0 >= n_nodes) return;              // wave-uniform: EXEC stays all-1s

    const int half = lane >> 4;               // 0: lanes 0-15, 1: lanes 16-31
    const int l16  = lane & 15;

    const int arow   = row0 + l16;
    const float scal = in_norm[arow];

    v8f acc0 = {};                            // cols 0..15
    v8f acc1 = {};                            // cols 16..31

#pragma unroll
    for (int k0 = 0; k0 < D; k0 += 4) {
        // A fragment (scaled by in_norm on the fly)
        const float* ap = agg + arow * D + k0 + half * 2;
        v2f a;
        a.x = ap[0] * scal;
        a.y = ap[1] * scal;

        // B fragments for both column tiles
        const float* wp = W + (k0 + half * 2) * D + l16;
        v2f b0; b0.x = wp[0];  b0.y = wp[D];       // K rows k0+half*2, +1 ; n = l16
        v2f b1; b1.x = wp[16]; b1.y = wp[D + 16];  // n = 16 + l16

        acc0 = __builtin_amdgcn_wmma_f32_16x16x4_f32(
                   false, a, false, b0, (short)0, acc0, false, false);
        acc1 = __builtin_amdgcn_wmma_f32_16x16x4_f32(
                   false, a, false, b1, (short)0, acc1, false, false);
    }

    const float bias0 = bias[l16];
    const float bias1 = bias[16 + l16];
#pragma unroll
    for (int i = 0; i < 8; ++i) {
        const int m = row0 + i + half * 8;
        out[m * D + l16]      = acc0[i] + bias0;
        out[m * D + 16 + l16] = acc1[i] + bias1;
    }
}

// ---------------------------------------------------------------------------
// launch
// ---------------------------------------------------------------------------
extern "C" void kernel_launch(void* const* d_in, const int* in_sizes, int n_in,
                              void* d_out, int out_size, void* d_ws, size_t ws_size,
                              hipStream_t stream) {
    const float* x   = (const float*)d_in[0];
    const float* W   = (const float*)d_in[1];
    const float* b   = (const float*)d_in[2];
    const int*   src = (const int*)d_in[3];
    const int*   dst = (const int*)d_in[4];
    float*       out = (float*)d_out;

    const int N = in_sizes[0] / D;   // 100000
    const int E = in_sizes[3];       // 6400000

    // workspace layout
    int*   deg_out  = (int*)d_ws;
    int*   deg_in   = deg_out + N;
    float* out_norm = (float*)(deg_in + N);
    float* in_norm  = out_norm + N;
    float* agg      = in_norm + N;   // N*D floats

    const int n_agg = N * D;

    gcn_init<<<(n_agg + 255) / 256, 256, 0, stream>>>(deg_out, deg_in, agg, N, n_agg);

    gcn_degree<<<(E + 255) / 256, 256, 0, stream>>>(src, dst, deg_out, deg_in, E);

    gcn_norm<<<(N + 255) / 256, 256, 0, stream>>>(deg_out, deg_in, out_norm, in_norm, N);

    // one wave (32 threads) per edge; 8 waves per 256-thread block
    gcn_scatter<<<(E + 7) / 8, 256, 0, stream>>>(x, src, dst, out_norm, agg, E);

    // one wave per 16-node row block
    const int tiles = (N + 15) / 16;
    gcn_gemm_wmma<<<(tiles + 7) / 8, 256, 0, stream>>>(agg, in_norm, W, b, out, N);
}